// ASIC_85624468013142
// MI455X (gfx1250) — compile-verified
//
#include <hip/hip_runtime.h>

// Problem constants (from reference)
namespace {
constexpr int kB = 16;      // batch
constexpr int kM = 4096;    // output width
constexpr int kN = 8192;    // state width
constexpr int kP = 64;      // 2^K patterns
constexpr int kL = 8;       // layers
// Tiling: each block produces kChunk final outputs; per layer the window is
// [n-2, n+3], so 8 layers need halo 16 left / 24 right.
constexpr int kChunk = 32;
constexpr int kHaloL = 16;
constexpr int kHaloR = 24;
constexpr int kRange = kChunk + kHaloL + kHaloR;  // 72 positions computed
constexpr int kWPad  = kRange + 8;                // state row stride (4 pad each side)
constexpr int kT = 96;                            // threads per block (3 waves)
}

#if defined(__HIP_DEVICE_COMPILE__) && __has_builtin(__builtin_amdgcn_tensor_load_to_lds) && __has_builtin(__builtin_amdgcn_s_wait_tensorcnt)
#define USE_TDM 1
#else
#define USE_TDM 0
#endif

typedef unsigned int v4u __attribute__((ext_vector_type(4)));
typedef int          v4i __attribute__((ext_vector_type(4)));
typedef int          v8i __attribute__((ext_vector_type(8)));

// Fallback / wrap-around path: cooperative coalesced load of one layer's
// 64 x kRange weight tile into LDS (row-major [p][j]).
__device__ __forceinline__ void coop_load_weights(float* swt, const float* __restrict__ tg,
                                                  int layer, int nstart, int tid) {
  const float* lsrc = tg + (long long)layer * ((long long)kP * kN);
  for (int idx = tid; idx < kP * kRange; idx += kT) {
    int p = idx / kRange;
    int j = idx - p * kRange;
    int n = nstart + j;
    if (n >= kN) n -= kN;
    swt[idx] = lsrc[(long long)p * kN + n];
  }
}

#if USE_TDM
// Issue one 2D TDM load: 64 rows (p) x kRange cols (n), row stride N, f32
// elements, packed contiguously into LDS at `swt`.
__device__ __forceinline__ void tdm_issue(const float* __restrict__ tileStart, float* swt) {
  unsigned long long ga = (unsigned long long)(size_t)tileStart;
  unsigned int ldsOff = (unsigned int)(size_t)(void*)swt;  // LDS byte offset
  v4u g0;
  g0.x = 1u;                                   // count=1, user mode, no gather
  g0.y = ldsOff;                               // lds_addr [63:32]
  g0.z = (unsigned int)ga;                     // global_addr [95:64]
  g0.w = (unsigned int)((ga >> 32) & 0x1FFFFFFull) | (2u << 30); // addr[56:32] | type=2
  v8i g1;
  g1[0] = 0x00020000;                          // data_size=2 (4B), no mask/pad/iterate
  g1[1] = (int)((unsigned)(kN & 0xFFFF) << 16);// tensor_dim0 lo16 at [63:48]
  g1[2] = (int)((unsigned)kP << 16);           // dim0 hi16=0 | tensor_dim1 lo16
  g1[3] = (int)((unsigned)kRange << 16);       // dim1 hi16=0 | tile_dim0
  g1[4] = kP;                                  // tile_dim1=64, tile_dim2=0
  g1[5] = kN;                                  // tensor_dim0_stride lo32
  g1[6] = 0;                                   // stride hi16 | dim1_stride lo16
  g1[7] = 0;
  v4i gz4 = {0, 0, 0, 0};
  v8i gz8 = {0, 0, 0, 0, 0, 0, 0, 0};
  __builtin_amdgcn_tensor_load_to_lds(g0, g1, gz4, gz4, gz8, 0);
}
#endif

__global__ __launch_bounds__(kT) void ASIC_fuzzy_lut_kernel(
    const float* __restrict__ x,    // [16][4096]
    const float* __restrict__ tg,   // [8][64][8192] raw gates (pre-sigmoid)
    float* __restrict__ out) {      // [16][4096]
  __shared__ float swt[2][kP * kRange];     // double-buffered weight tiles
  __shared__ float sst[2][kB * kWPad];      // ping-pong state

  const int tid = threadIdx.x;
  const int a = blockIdx.x * kChunk;                 // first final-output position
  const int nstart = (a - kHaloL + kN) % kN;         // first staged position (mod N)
  const bool wraps = (nstart + kRange > kN);         // tile crosses the circular seam

#if USE_TDM
  // Pipeline prologue: start DMA of layer 0's weight tile immediately.
  if (!wraps && tid < 32) {
    tdm_issue(tg + nstart, swt[0]);
  }
#endif

  // Zero state buffers (covers pad columns), then set even positions from x.
  for (int i = tid; i < kB * kWPad; i += kT) { sst[0][i] = 0.0f; sst[1][i] = 0.0f; }
  __syncthreads();
  if (tid < kRange) {
    int n = nstart + tid;
    if (n >= kN) n -= kN;
    if ((n & 1) == 0) {
      int m = n >> 1;
      #pragma unroll
      for (int b = 0; b < kB; ++b) sst[0][b * kWPad + 4 + tid] = x[b * kM + m];
    }
  }

  int cur = 0;
  for (int layer = 0; layer < kL; ++layer) {
    const int wb = layer & 1;  // weight buffer holding this layer's tile
#if USE_TDM
    if (!wraps) {
      if (tid < 32) {
        if (layer + 1 < kL) {
          // Kick off next layer's DMA into the other buffer, then wait only
          // for the older (this layer's) transfer: TENSORcnt is in-order,
          // so cnt<=1 means layer `layer` has landed in LDS.
          const long long goffN = (long long)(layer + 1) * ((long long)kP * kN) + nstart;
          tdm_issue(tg + goffN, swt[wb ^ 1]);
          __builtin_amdgcn_s_wait_tensorcnt(1);
        } else {
          __builtin_amdgcn_s_wait_tensorcnt(0);
        }
      }
    } else {
      coop_load_weights(swt[wb], tg, layer, nstart, tid);
    }
#else
    coop_load_weights(swt[wb], tg, layer, nstart, tid);
#endif
    __syncthreads();

    if (tid < kRange) {
      // Load this lane's 64 gates, apply sigmoid once; reused for all 16 b.
      float w[kP];
      #pragma unroll
      for (int p = 0; p < kP; ++p) {
        float g = swt[wb][p * kRange + tid];
        w[p] = __builtin_amdgcn_rcpf(1.0f + __expf(-g));
      }

      const float* st  = sst[cur];
      float*       stn = sst[cur ^ 1];
      for (int b = 0; b < kB; ++b) {
        const float* row = st + b * kWPad + tid;  // center column is tid+4
        float v0 = row[2], v1 = row[3], v2 = row[4];
        float v3 = row[5], v4 = row[6], v5 = row[7];
        // Multilinear interpolation of the 64-entry table at (v0..v5).
        // bit_k(p) = (p >> (5-k)) & 1, so LSB pairs with v5: contract LSB-first.
        float u[32];
        #pragma unroll
        for (int q = 0; q < 32; ++q) u[q] = fmaf(v5, w[2*q+1] - w[2*q], w[2*q]);
        #pragma unroll
        for (int q = 0; q < 16; ++q) u[q] = fmaf(v4, u[2*q+1] - u[2*q], u[2*q]);
        #pragma unroll
        for (int q = 0; q < 8;  ++q) u[q] = fmaf(v3, u[2*q+1] - u[2*q], u[2*q]);
        #pragma unroll
        for (int q = 0; q < 4;  ++q) u[q] = fmaf(v2, u[2*q+1] - u[2*q], u[2*q]);
        float t0 = fmaf(v1, u[1] - u[0], u[0]);
        float t1 = fmaf(v1, u[3] - u[2], u[2]);
        float r  = fmaf(v0, t1 - t0, t0);
        r = fminf(fmaxf(r, 0.0f), 1.0f);
        stn[b * kWPad + 4 + tid] = r;
      }
    }
    __syncthreads();
    cur ^= 1;
  }

  // Emit even positions of the valid center region.
  if (tid >= kHaloL && tid < kHaloL + kChunk) {
    int n = a + (tid - kHaloL);
    if ((n & 1) == 0) {
      int m = n >> 1;
      #pragma unroll
      for (int b = 0; b < kB; ++b)
        out[b * kM + m] = sst[cur][b * kWPad + 4 + tid];
    }
  }
}

extern "C" void kernel_launch(void* const* d_in, const int* in_sizes, int n_in,
                              void* d_out, int out_size, void* d_ws, size_t ws_size,
                              hipStream_t stream) {
  (void)in_sizes; (void)n_in; (void)d_ws; (void)ws_size; (void)out_size;
  const float* x  = (const float*)d_in[0];   // (16, 4096) f32
  const float* tg = (const float*)d_in[1];   // (8, 64, 8192) f32
  float* out = (float*)d_out;                // (16, 4096) f32
  dim3 grid(kN / kChunk);   // 256 blocks
  dim3 block(kT);           // 96 threads (3 waves)
  ASIC_fuzzy_lut_kernel<<<grid, block, 0, stream>>>(x, tg, out);
}